// _DeepJDOTAligner_81011673137740
// MI455X (gfx1250) — compile-verified
//
#include <hip/hip_runtime.h>
#include <hip/hip_bf16.h>
#include <stdint.h>

#define NS 4096
#define NT 4096
#define DD 512
#define CNUM 100
#define EPS16 1e-16f
// 1/SINKHORN_REG = 10

typedef __attribute__((ext_vector_type(2))) float v2f;
typedef __attribute__((ext_vector_type(8))) float v8f;

// ---------------------------------------------------------------- init
__global__ void k_init(float* u, float* v, unsigned int* maxbits, float* out) {
    int t = blockIdx.x * blockDim.x + threadIdx.x;
    if (t < NS) u[t] = 1.0f / NS;
    if (t < NT) v[t] = 1.0f / NT;
    if (t == 0) { *maxbits = 0u; out[0] = 0.0f; }
}

// ------------------------------------------------- log-sum-exp per target row
__global__ __launch_bounds__(256) void k_lse(const float* __restrict__ logits,
                                             float* __restrict__ lse) {
    int wid  = (blockIdx.x * blockDim.x + threadIdx.x) >> 5;   // row j
    int lane = threadIdx.x & 31;
    if (wid >= NT) return;
    const float* row = logits + (size_t)wid * CNUM;
    float mx = -3.4e38f;
    for (int c = lane; c < CNUM; c += 32) mx = fmaxf(mx, row[c]);
    #pragma unroll
    for (int off = 16; off; off >>= 1) mx = fmaxf(mx, __shfl_xor(mx, off, 32));
    float s = 0.0f;
    for (int c = lane; c < CNUM; c += 32) s += __expf(row[c] - mx);
    #pragma unroll
    for (int off = 16; off; off >>= 1) s += __shfl_xor(s, off, 32);
    if (lane == 0) lse[wid] = mx + __logf(s);
}

// ------------------------------------------------- CE table [C][NT]
__global__ void k_cetab(const float* __restrict__ logits, const float* __restrict__ lse,
                        float* __restrict__ ce_tab) {
    int t = blockIdx.x * blockDim.x + threadIdx.x;     // t = c*NT + j
    if (t >= CNUM * NT) return;
    int c = t >> 12;          // /NT
    int j = t & (NT - 1);
    ce_tab[t] = lse[j] - logits[(size_t)j * CNUM + c];
}

// ------------------------------------------------- row sum of squares
__global__ __launch_bounds__(256) void k_rownorm(const float* __restrict__ X,
                                                 float* __restrict__ out, int nrows) {
    int wid  = (blockIdx.x * blockDim.x + threadIdx.x) >> 5;
    int lane = threadIdx.x & 31;
    if (wid >= nrows) return;
    const float4* row = (const float4*)(X + (size_t)wid * DD);
    float acc = 0.0f;
    #pragma unroll
    for (int c = lane; c < DD / 4; c += 32) {
        float4 t = row[c];
        acc += t.x * t.x + t.y * t.y + t.z * t.z + t.w * t.w;
    }
    #pragma unroll
    for (int off = 16; off; off >>= 1) acc += __shfl_xor(acc, off, 32);
    if (lane == 0) out[wid] = acc;
}

// ------------------------------------------------- f32 WMMA GEMM + sq epilogue + global max
// block = 128 threads = 4 waves in 2x2; each wave computes a 64x64 tile (4x4 WMMA tiles).
__global__ __launch_bounds__(128) void k_gemm_sq(const float* __restrict__ Fs,
                                                 const float* __restrict__ Ft,
                                                 const float* __restrict__ rs,
                                                 const float* __restrict__ rt,
                                                 float* __restrict__ S,
                                                 unsigned int* __restrict__ maxbits) {
    const int lane = threadIdx.x & 31;
    const int wave = threadIdx.x >> 5;    // 0..3
    const int wm   = wave >> 1;
    const int wn   = wave & 1;
    const int Mbase = blockIdx.y * 128 + wm * 64;
    const int Nbase = blockIdx.x * 128 + wn * 64;
    const int kh  = lane >> 4;            // half-wave: selects K pair
    const int l15 = lane & 15;

    v8f acc[4][4];
    #pragma unroll
    for (int mt = 0; mt < 4; mt++)
        #pragma unroll
        for (int nt = 0; nt < 4; nt++)
            acc[mt][nt] = (v8f){0.f,0.f,0.f,0.f,0.f,0.f,0.f,0.f};

    const float* arow[4];
    const float* brow[4];
    #pragma unroll
    for (int mt = 0; mt < 4; mt++) arow[mt] = Fs + (size_t)(Mbase + mt * 16 + l15) * DD;
    #pragma unroll
    for (int nt = 0; nt < 4; nt++) brow[nt] = Ft + (size_t)(Nbase + nt * 16 + l15) * DD;

    for (int kk = 0; kk < DD; kk += 4) {
        const int ko = kk + 2 * kh;       // lanes 0-15: K={kk,kk+1}; 16-31: {kk+2,kk+3}
        v2f a[4], b[4];
        #pragma unroll
        for (int mt = 0; mt < 4; mt++) a[mt] = *(const v2f*)(arow[mt] + ko);
        #pragma unroll
        for (int nt = 0; nt < 4; nt++) b[nt] = *(const v2f*)(brow[nt] + ko);
        #pragma unroll
        for (int mt = 0; mt < 4; mt++)
            #pragma unroll
            for (int nt = 0; nt < 4; nt++)
                acc[mt][nt] = __builtin_amdgcn_wmma_f32_16x16x4_f32(
                    false, a[mt], false, b[nt], (short)0, acc[mt][nt], false, false);
    }

    // epilogue: sq = rs + rt - 2*G, clamp, store, track max
    float mx = 0.0f;
    #pragma unroll
    for (int mt = 0; mt < 4; mt++) {
        const int i0 = Mbase + mt * 16 + kh * 8;     // rows for VGPR r = i0 + r
        float rsv[8];
        #pragma unroll
        for (int r = 0; r < 8; r++) rsv[r] = rs[i0 + r];
        #pragma unroll
        for (int nt = 0; nt < 4; nt++) {
            const int j = Nbase + nt * 16 + l15;
            const float rtj = rt[j];
            #pragma unroll
            for (int r = 0; r < 8; r++) {
                float sq = rsv[r] + rtj - 2.0f * acc[mt][nt][r];
                sq = fmaxf(sq, 0.0f);
                mx = fmaxf(mx, sq);
                S[(size_t)(i0 + r) * NT + j] = sq;
            }
        }
    }
    #pragma unroll
    for (int off = 16; off; off >>= 1) mx = fmaxf(mx, __shfl_xor(mx, off, 32));
    __shared__ float smax[4];
    if (lane == 0) smax[wave] = mx;
    __syncthreads();
    if (threadIdx.x == 0) {
        float m = fmaxf(fmaxf(smax[0], smax[1]), fmaxf(smax[2], smax[3]));
        atomicMax(maxbits, __float_as_uint(m));   // valid: all values >= 0
    }
}

// ------------------------------------------------- K = exp(-10*M) in place; also emit K^T
__global__ __launch_bounds__(256) void k_expk_tr(float* __restrict__ S,
                                                 float* __restrict__ KT,
                                                 const float* __restrict__ ce_tab,
                                                 const int* __restrict__ lab,
                                                 const unsigned int* __restrict__ maxbits) {
    __shared__ float tile[32][33];
    const float mval = __uint_as_float(*maxbits);
    const float inv  = 1.0f / fmaxf(mval, EPS16);
    const int jb = blockIdx.x * 32, ib = blockIdx.y * 32;
    const int tx = threadIdx.x & 31;        // j within tile
    const int ty = threadIdx.x >> 5;        // 0..7
    #pragma unroll
    for (int r = 0; r < 4; r++) {
        const int li = ty + r * 8;
        const int i = ib + li, j = jb + tx;
        float sq = S[(size_t)i * NT + j];
        float m  = sq * inv + ce_tab[(size_t)lab[i] * NT + j];
        float K  = __expf(-10.0f * m);
        S[(size_t)i * NT + j] = K;
        tile[li][tx] = K;
    }
    __syncthreads();
    #pragma unroll
    for (int r = 0; r < 4; r++) {
        const int lj = ty + r * 8;
        KT[(size_t)(jb + lj) * NS + (ib + tx)] = tile[tx][lj];
    }
}

// ------------------------------------------------- out[i] = scale / max((A x)[i], eps)
__global__ __launch_bounds__(256) void k_matvec(const float* __restrict__ A,
                                                const float* __restrict__ x,
                                                float* __restrict__ out, float scale) {
    const int wid  = (blockIdx.x * blockDim.x + threadIdx.x) >> 5;  // row
    const int lane = threadIdx.x & 31;
    const float4* Ar = (const float4*)(A + (size_t)wid * 4096);
    const float4* xv = (const float4*)x;
    float acc = 0.0f;
    for (int c = lane; c < 1024; c += 32) {
        float4 a4 = Ar[c];
        float4 x4 = xv[c];
        acc += a4.x * x4.x + a4.y * x4.y + a4.z * x4.z + a4.w * x4.w;
    }
    #pragma unroll
    for (int off = 16; off; off >>= 1) acc += __shfl_xor(acc, off, 32);
    if (lane == 0) out[wid] = scale / fmaxf(acc, EPS16);
}

// ------------------------------------------------- loss = sum u_i K_ij v_j * (-0.1 ln K_ij)
__global__ __launch_bounds__(256) void k_loss(const float* __restrict__ K,
                                              const float* __restrict__ u,
                                              const float* __restrict__ v,
                                              float* __restrict__ out) {
    const size_t b0 = (size_t)blockIdx.x * 2048;
    float acc = 0.0f;
    #pragma unroll
    for (int t = 0; t < 8; t++) {
        const size_t e = b0 + (size_t)t * 256 + threadIdx.x;
        const int i = (int)(e >> 12);
        const int j = (int)(e & 4095);
        float k = K[e];
        if (k > 0.0f) {                       // K==0  =>  gamma==0  =>  contributes 0
            float m = -0.1f * __logf(k);
            acc += u[i] * k * v[j] * m;
        }
    }
    #pragma unroll
    for (int off = 16; off; off >>= 1) acc += __shfl_xor(acc, off, 32);
    __shared__ float sred[8];
    const int lane = threadIdx.x & 31, w = threadIdx.x >> 5;
    if (lane == 0) sred[w] = acc;
    __syncthreads();
    if (threadIdx.x == 0) {
        float s = 0.0f;
        #pragma unroll
        for (int q = 0; q < 8; q++) s += sred[q];
        atomicAdd(out, s);
    }
}

// ----------------------------------------------------------------------------
extern "C" void kernel_launch(void* const* d_in, const int* in_sizes, int n_in,
                              void* d_out, int out_size, void* d_ws, size_t ws_size,
                              hipStream_t stream) {
    const int*   lab = (const int*)d_in[0];                 // [NS]
    const float* lg  = (const float*)d_in[1];               // [NT, C]
    const float* Fs  = (const float*)d_in[2];               // [NS, D]
    const float* Ft  = (const float*)d_in[3];               // [NT, D]
    float* out = (float*)d_out;

    char* ws = (char*)d_ws;
    size_t off = 0;
    float* S   = (float*)(ws + off); off += (size_t)NS * NT * 4;   // sq, then K (in place)
    float* KT  = (float*)(ws + off); off += (size_t)NT * NS * 4;   // K transposed
    float* ce  = (float*)(ws + off); off += (size_t)CNUM * NT * 4; // CE table [C][NT]
    float* lse = (float*)(ws + off); off += (size_t)NT * 4;
    float* rs  = (float*)(ws + off); off += (size_t)NS * 4;
    float* rt  = (float*)(ws + off); off += (size_t)NT * 4;
    float* u   = (float*)(ws + off); off += (size_t)NS * 4;
    float* v   = (float*)(ws + off); off += (size_t)NT * 4;
    unsigned int* maxbits = (unsigned int*)(ws + off); off += 256;

    k_init<<<16, 256, 0, stream>>>(u, v, maxbits, out);
    k_lse<<<NT / 8, 256, 0, stream>>>(lg, lse);
    k_cetab<<<(CNUM * NT + 255) / 256, 256, 0, stream>>>(lg, lse, ce);
    k_rownorm<<<NS / 8, 256, 0, stream>>>(Fs, rs, NS);
    k_rownorm<<<NT / 8, 256, 0, stream>>>(Ft, rt, NT);

    k_gemm_sq<<<dim3(NT / 128, NS / 128), 128, 0, stream>>>(Fs, Ft, rs, rt, S, maxbits);
    k_expk_tr<<<dim3(NT / 32, NS / 32), 256, 0, stream>>>(S, KT, ce, lab, maxbits);

    for (int it = 0; it < 100; it++) {
        k_matvec<<<NS / 8, 256, 0, stream>>>(S,  v, u, 1.0f / NS);  // u = a/max(Kv)
        k_matvec<<<NT / 8, 256, 0, stream>>>(KT, u, v, 1.0f / NT);  // v = b/max(K^T u)
    }

    k_loss<<<(int)(((size_t)NS * NT) / 2048), 256, 0, stream>>>(S, u, v, out);
}